// sparse_low_precision_linear_78640851190261
// MI455X (gfx1250) — compile-verified
//
#include <hip/hip_runtime.h>

// MI455X / gfx1250, wave32.
// out^T = W^T (2:4-sparse int4, SWMMAC A operand) x x^T (f16, B operand)
// via V_SWMMAC_F32_16X16X64_F16. A tiles dequantized once per workgroup and
// shared through LDS; magic-number fp16 dequant (0x6400|q -> 1024+q).

typedef __attribute__((ext_vector_type(32))) _Float16 v32h;
typedef __attribute__((ext_vector_type(16))) _Float16 v16h;
typedef __attribute__((ext_vector_type(8)))  _Float16 v8h;
typedef __attribute__((ext_vector_type(2)))  _Float16 h2;
typedef __attribute__((ext_vector_type(8)))  float    v8f;

#define MDIM 4096
#define KDIM 4096
#define NDIM 8192
#define GROUPSIZE 128

// Pattern p -> sparse index nibble: idx0 in [1:0], idx1 in [3:2] (idx0 < idx1)
// (0,1)->4 (0,2)->8 (0,3)->C (1,2)->9 (1,3)->D (2,3)->E
#define ILUT 0x00ED9C84u

// ---------------------------------------------------------------------------
// WG = 8 waves = 2(N-half) x 4(M-quarter) over a 128x128 out^T tile.
// Each wave: consumes 4 A blocks (its 64 N cols) x 2 B blocks (32 M rows),
// but PRODUCES only 1 A block (dequant) into LDS; 4 waves/half share.
// K step = 64 (one SWMMAC per 16x16 block pair). Double-buffered LDS.
// ---------------------------------------------------------------------------
__global__ __launch_bounds__(256)
void sq24_swmmac_kernel(const _Float16* __restrict__ x,
                        const int*      __restrict__ qweight,
                        const int*      __restrict__ meta,
                        const _Float16* __restrict__ scale,
                        _Float16*       __restrict__ out)
{
    const int lane   = threadIdx.x & 31;
    const int wave   = threadIdx.x >> 5;
    const int l16    = lane & 15;
    const int half16 = lane >> 4;
    const int h      = wave >> 2;     // N-half of this wave (0..1)
    const int qr     = wave & 3;      // M-quarter; also the A block it produces

    const int nBase = blockIdx.x * 128 + h * 64;         // wave's 64 out cols
    const int mBase = blockIdx.y * 128 + qr * 32;        // wave's 32 out rows
    const int ncol  = nBase + qr * 16 + l16;             // column this wave dequants

    // LDS: [buf][half][block][lane] -> 16 packed f16 values + 1 index word
    __shared__ _Float16 lA[2][2][4][32][16];             // 16 KB
    __shared__ int      lI[2][2][4][32];                 // 2 KB

    // Per-lane producer base pointers (advanced by constant strides per iter).
    const int* qPtr    = qweight + (size_t)(half16 * 16) * NDIM + ncol; // k row
    const int* mValPtr = meta    + (size_t)(half16 * 4)  * NDIM + ncol; // value groups
    const int* mIdxPtr = meta    + (size_t)(half16 * 8)  * NDIM + ncol; // index groups
    const _Float16* scPtr = scale + ncol;

    v8f acc[4][2] = {};

    for (int k0 = 0; k0 < KDIM; k0 += 64) {
        const int buf = (k0 >> 6) & 1;

        // ================= producer: dequant A block 'qr' of half 'h' ========
        {
            const _Float16 sh = scPtr[(size_t)((unsigned)k0 >> 7) * NDIM];
            const h2 s2 = { sh, sh };
            const h2 b2 = { (_Float16)(sh * (_Float16)-1032.0f),
                            (_Float16)(sh * (_Float16)-1032.0f) };

            // sparse-index VGPR: lanes 0-15 -> groups +0..7, lanes 16-31 -> +8..15
            unsigned idxbits = 0;
            #pragma unroll
            for (int g = 0; g < 8; ++g) {
                const int m = mIdxPtr[(size_t)g * NDIM];
                idxbits |= ((ILUT >> (m * 4)) & 0xFu) << (g * 4);
            }

            // stored nonzeros: slot (c*8 + 2u + sel) <- group (c*8? no: chunk c)
            // chunk c in {0,1}: stored cols 16c + half16*8 + 2u+sel
            //   -> group offset c*8 + u (relative to this lane's value base),
            //      k row offset c*32 + 4u (relative to qPtr).
            v16h av;
            #pragma unroll
            for (int u = 0; u < 4; ++u) {
                #pragma unroll
                for (int c = 0; c < 2; ++c) {
                    const int mv = mValPtr[(size_t)(u + c * 8) * NDIM];
                    const unsigned r = ILUT >> (mv * 4);
                    const int* qg = qPtr + (size_t)(u * 4 + c * 32) * NDIM;
                    const int q0 = qg[(size_t)(r & 3u) * NDIM];
                    const int q1 = qg[(size_t)((r >> 2) & 3u) * NDIM];
                    const unsigned packed =
                        ((unsigned)q0 | ((unsigned)q1 << 16)) | 0x64006400u;
                    h2 hv = __builtin_bit_cast(h2, packed);     // {1024+q0, 1024+q1}
                    h2 w  = __builtin_elementwise_fma(hv, s2, b2); // (q-8)*s
                    av[c * 8 + 2 * u]     = w.x;
                    av[c * 8 + 2 * u + 1] = w.y;
                }
            }

            *(v16h*)&lA[buf][h][qr][lane][0] = av;
            lI[buf][h][qr][lane] = (int)idxbits;
        }

        // ---- B operands: x^T 64x16 dense f16 (per-wave, from global) ----
        v32h b[2];
        #pragma unroll
        for (int mb = 0; mb < 2; ++mb) {
            const int m = mBase + mb * 16 + l16;
            const _Float16* xr = x + (size_t)m * KDIM + k0 + half16 * 16;
            union { v32h v; v8h hh[4]; } u;
            u.hh[0] = *(const v8h*)(xr);
            u.hh[1] = *(const v8h*)(xr + 8);
            u.hh[2] = *(const v8h*)(xr + 32);
            u.hh[3] = *(const v8h*)(xr + 40);
            b[mb] = u.v;
        }

        // L2 prefetch of next K-slab of qweight (weights are L2-resident)
        if (k0 + 64 < KDIM)
            __builtin_prefetch(&qweight[(size_t)(k0 + 64) * NDIM + nBase + lane], 0, 1);

        __syncthreads();   // A tiles of this buffer are complete

        // ================= consumer: 4 A blocks x 2 B blocks ================
        #pragma unroll
        for (int ab = 0; ab < 4; ++ab) {
            const v16h a  = *(const v16h*)&lA[buf][h][ab][lane][0];
            const int  ai = lI[buf][h][ab][lane];
            acc[ab][0] = __builtin_amdgcn_swmmac_f32_16x16x64_f16(
                false, a, false, b[0], acc[ab][0], ai, false, false);
            acc[ab][1] = __builtin_amdgcn_swmmac_f32_16x16x64_f16(
                false, a, false, b[1], acc[ab][1], ai, false, false);
        }

        // advance producer pointers by one 64-K slab
        qPtr    += (size_t)64 * NDIM;
        mValPtr += (size_t)16 * NDIM;
        mIdxPtr += (size_t)16 * NDIM;
    }

    // ---- store: D row r -> out col n (contiguous over r) => b128 stores ----
    #pragma unroll
    for (int ab = 0; ab < 4; ++ab) {
        #pragma unroll
        for (int mb = 0; mb < 2; ++mb) {
            const int m = mBase + mb * 16 + l16;
            const int n = nBase + ab * 16 + half16 * 8;
            v8h o;
            #pragma unroll
            for (int r = 0; r < 8; ++r)
                o[r] = (_Float16)acc[ab][mb][r];
            *(v8h*)(out + (size_t)m * NDIM + n) = o;
        }
    }
}

extern "C" void kernel_launch(void* const* d_in, const int* in_sizes, int n_in,
                              void* d_out, int out_size, void* d_ws, size_t ws_size,
                              hipStream_t stream) {
    (void)in_sizes; (void)n_in; (void)d_ws; (void)ws_size; (void)out_size;
    const _Float16* x       = (const _Float16*)d_in[0];
    const int*      qweight = (const int*)d_in[1];
    const int*      meta    = (const int*)d_in[2];
    const _Float16* scale   = (const _Float16*)d_in[3];
    _Float16*       out     = (_Float16*)d_out;

    dim3 grid(NDIM / 128, MDIM / 128);
    dim3 block(256);
    sq24_swmmac_kernel<<<grid, block, 0, stream>>>(x, qweight, meta, scale, out);
}